// GemmaSlidingLayer_34832184770694
// MI455X (gfx1250) — compile-verified
//
#include <hip/hip_runtime.h>
#include <math.h>

// ---------------- model constants ----------------
constexpr int   kD    = 2816;
constexpr int   kDD   = 2112;
constexpr int   kFF   = 704;
constexpr int   kG    = 16;
constexpr int   kP    = 4;
constexpr int   kH    = 16;
constexpr int   kNKV  = 8;
constexpr int   kDH   = 256;
constexpr int   kNE   = 64;   // P*G
constexpr int   kTOPK = 8;
constexpr int   kCTX  = 4096;
constexpr float kEPS  = 1e-6f;

// WMMA f32 16x16x4: A=16x4 (2 VGPR), B=4x16 (2 VGPR), C/D=16x16 (8 VGPR)
typedef float v2f __attribute__((ext_vector_type(2)));
typedef float v8f __attribute__((ext_vector_type(8)));

// ---------------- reductions ----------------
__device__ __forceinline__ float wave_sum(float v) {
#pragma unroll
  for (int o = 16; o > 0; o >>= 1) v += __shfl_xor(v, o, 32);
  return v;
}

__device__ __forceinline__ float block_sum_256(float v) {
  __shared__ float sh[8];
  __shared__ float tot;
  v = wave_sum(v);
  if ((threadIdx.x & 31) == 0) sh[threadIdx.x >> 5] = v;
  __syncthreads();
  if (threadIdx.x == 0) {
    float s = 0.f;
#pragma unroll
    for (int i = 0; i < 8; ++i) s += sh[i];
    tot = s;
  }
  __syncthreads();
  float r = tot;
  __syncthreads();
  return r;
}

__device__ __forceinline__ float block_max_256(float v) {
  __shared__ float shm[8];
  __shared__ float tm;
#pragma unroll
  for (int o = 16; o > 0; o >>= 1) v = fmaxf(v, __shfl_xor(v, o, 32));
  if ((threadIdx.x & 31) == 0) shm[threadIdx.x >> 5] = v;
  __syncthreads();
  if (threadIdx.x == 0) {
    float m = shm[0];
#pragma unroll
    for (int i = 1; i < 8; ++i) m = fmaxf(m, shm[i]);
    tm = m;
  }
  __syncthreads();
  float r = tm;
  __syncthreads();
  return r;
}

__device__ __forceinline__ float gelu_tanh(float x) {
  float x3 = x * x * x;
  return 0.5f * x * (1.0f + tanhf(0.7978845608028654f * (x + 0.044715f * x3)));
}

// ---------------- rmsnorm family (single block, 256 threads) ----------------
__global__ void rmsnorm_kernel(const float* __restrict__ in, const float* __restrict__ w,
                               float* __restrict__ out, int n) {
  float ss = 0.f;
  for (int i = threadIdx.x; i < n; i += 256) { float v = in[i]; ss = fmaf(v, v, ss); }
  float tot = block_sum_256(ss);
  float r = rsqrtf(tot / (float)n + kEPS);
  for (int i = threadIdx.x; i < n; i += 256) out[i] = in[i] * r * (1.0f + w[i]);
}

__global__ void add_rmsnorm_kernel(const float* __restrict__ res, const float* __restrict__ in,
                                   const float* __restrict__ w, float* __restrict__ out, int n) {
  float ss = 0.f;
  for (int i = threadIdx.x; i < n; i += 256) { float v = in[i]; ss = fmaf(v, v, ss); }
  float tot = block_sum_256(ss);
  float r = rsqrtf(tot / (float)n + kEPS);
  for (int i = threadIdx.x; i < n; i += 256) out[i] = res[i] + in[i] * r * (1.0f + w[i]);
}

__global__ void router_pre_kernel(const float* __restrict__ in, const float* __restrict__ scale,
                                  float* __restrict__ out, int n) {
  float ss = 0.f;
  for (int i = threadIdx.x; i < n; i += 256) { float v = in[i]; ss = fmaf(v, v, ss); }
  float tot = block_sum_256(ss);
  float r = rsqrtf(tot / (float)n + kEPS) * rsqrtf((float)kD);
  for (int i = threadIdx.x; i < n; i += 256) out[i] = in[i] * r * scale[i];
}

__global__ void final_kernel(const float* __restrict__ x2, const float* __restrict__ h1,
                             const float* __restrict__ h2, const float* __restrict__ w,
                             const float* __restrict__ layer_scalar, float* __restrict__ out) {
  float ss = 0.f;
  for (int i = threadIdx.x; i < kD; i += 256) {
    float v = h1[i] + h2[i];
    ss = fmaf(v, v, ss);
  }
  float tot = block_sum_256(ss);
  float r = rsqrtf(tot / (float)kD + kEPS);
  float ls = layer_scalar[0];
  for (int i = threadIdx.x; i < kD; i += 256) {
    float s = h1[i] + h2[i];
    out[i] = (x2[i] + s * r * (1.0f + w[i])) * ls;
  }
}

// ---------------- streaming GEMV (one wave per row, float4 loads) ----------------
__global__ void gemv_kernel(const float* __restrict__ W, const float* __restrict__ x,
                            float* __restrict__ y, int nrows, int d4) {
  int row  = blockIdx.x * 8 + (threadIdx.x >> 5);
  int lane = threadIdx.x & 31;
  if (row >= nrows) return;
  const float4* Wr = (const float4*)W + (size_t)row * d4;
  const float4* X4 = (const float4*)x;
  float acc = 0.f;
  for (int i = lane; i < d4; i += 32) {
    float4 w = Wr[i];
    float4 v = X4[i];
    acc = fmaf(w.x, v.x, fmaf(w.y, v.y, fmaf(w.z, v.z, fmaf(w.w, v.w, acc))));
  }
  acc = wave_sum(acc);
  if (lane == 0) y[row] = acc;
}

// fused gate/up GEMV; optionally skip rows whose expert router-weight is zero
__global__ void gateup_kernel(const float* __restrict__ Wg, const float* __restrict__ Wu,
                              const float* __restrict__ x, float* __restrict__ t,
                              const float* __restrict__ rwp, int rw_div,
                              int nrows, int d4) {
  int row  = blockIdx.x * 8 + (threadIdx.x >> 5);
  int lane = threadIdx.x & 31;
  if (row >= nrows) return;
  if (rwp != nullptr && rwp[row / rw_div] == 0.0f) return;  // inactive expert: skip
  const float4* G4 = (const float4*)Wg + (size_t)row * d4;
  const float4* U4 = (const float4*)Wu + (size_t)row * d4;
  const float4* X4 = (const float4*)x;
  float g = 0.f, u = 0.f;
  for (int i = lane; i < d4; i += 32) {
    float4 xv = X4[i];
    float4 gv = G4[i];
    float4 uv = U4[i];
    g = fmaf(gv.x, xv.x, fmaf(gv.y, xv.y, fmaf(gv.z, xv.z, fmaf(gv.w, xv.w, g))));
    u = fmaf(uv.x, xv.x, fmaf(uv.y, xv.y, fmaf(uv.z, xv.z, fmaf(uv.w, xv.w, u))));
  }
  g = wave_sum(g);
  u = wave_sum(u);
  if (lane == 0) t[row] = gelu_tanh(g) * u;
}

// moe[d] = sum_pg rw[pg] * dot(hh[pg,:], pack_down[pg*D+d, :]); skips rw==0 experts
__global__ void pack_down_kernel(const float* __restrict__ Wd, const float* __restrict__ hh,
                                 const float* __restrict__ rw, float* __restrict__ moe) {
  int d    = blockIdx.x * 8 + (threadIdx.x >> 5);
  int lane = threadIdx.x & 31;
  if (d >= kD) return;
  float acc = 0.f;
  for (int pg = 0; pg < kNE; ++pg) {
    float w = rw[pg];
    if (w == 0.0f) continue;                      // uniform per-wave branch
    const float4* Wr = (const float4*)Wd + ((size_t)pg * kD + d) * (kFF / 4);
    const float4* Hh = (const float4*)(hh + (size_t)pg * kFF);
    float dot = 0.f;
    for (int i = lane; i < kFF / 4; i += 32) {
      float4 a = Wr[i];
      float4 b = Hh[i];
      dot = fmaf(a.x, b.x, fmaf(a.y, b.y, fmaf(a.z, b.z, fmaf(a.w, b.w, dot))));
    }
    acc = fmaf(w, dot, acc);
  }
  acc = wave_sum(acc);
  if (lane == 0) moe[d] = acc;
}

// ---------------- per-head norm + rope ----------------
// blocks 0..15: q heads (q_norm + rope); 16..23: k heads (k_norm + rope); 24..31: v heads (plain norm)
__global__ void qkv_norm_rope_kernel(const float* __restrict__ qraw, const float* __restrict__ kraw,
                                     const float* __restrict__ vraw,
                                     const float* __restrict__ qnw, const float* __restrict__ knw,
                                     const float* __restrict__ cosv, const float* __restrict__ sinv,
                                     float* __restrict__ qn, float* __restrict__ kn,
                                     float* __restrict__ vn) {
  int b = blockIdx.x;
  int t = threadIdx.x;  // 256 == kDH
  const float* src;
  float* dst;
  const float* w;
  bool do_rope;
  if (b < kH)            { src = qraw + (size_t)b * kDH;          dst = qn + (size_t)b * kDH;          w = qnw;     do_rope = true;  }
  else if (b < kH + kNKV){ src = kraw + (size_t)(b - kH) * kDH;   dst = kn + (size_t)(b - kH) * kDH;   w = knw;     do_rope = true;  }
  else                   { src = vraw + (size_t)(b - kH - kNKV) * kDH; dst = vn + (size_t)(b - kH - kNKV) * kDH; w = nullptr; do_rope = false; }
  float v = src[t];
  float tot = block_sum_256(v * v);
  float r = rsqrtf(tot / (float)kDH + kEPS);
  float normed = v * r * (w ? (1.0f + w[t]) : 1.0f);
  __shared__ float sh[kDH];
  sh[t] = normed;
  __syncthreads();
  float outv;
  if (do_rope) {
    float rot = (t < kDH / 2) ? -sh[t + kDH / 2] : sh[t - kDH / 2];
    outv = normed * cosv[t] + rot * sinv[t];
  } else {
    outv = normed;
  }
  dst[t] = outv;
}

// ---------------- KV cache write-out: out = cache + mask[pos] * kv[head,:] ----------------
__global__ void kv_update_kernel(const float* __restrict__ cache, const float* __restrict__ kv,
                                 const float* __restrict__ mask, float* __restrict__ outc) {
  size_t i = (size_t)blockIdx.x * 256 + threadIdx.x;  // float4 index, total NKV*CTX*DH/4
  int d4      = (int)(i & 63);        // DH/4 = 64
  size_t rest = i >> 6;               // h*CTX + pos
  int pos     = (int)(rest & (kCTX - 1));
  int hidx    = (int)(rest >> 12);
  float4 v = ((const float4*)cache)[i];
  float m = mask[pos];
  if (m != 0.0f) {
    float4 a = ((const float4*)kv)[(size_t)hidx * (kDH / 4) + d4];
    v.x = fmaf(m, a.x, v.x); v.y = fmaf(m, a.y, v.y);
    v.z = fmaf(m, a.z, v.z); v.w = fmaf(m, a.w, v.w);
  }
  ((float4*)outc)[i] = v;
}

// ---------------- attention scores via V_WMMA_F32_16X16X4_F32 ----------------
// wave = (kv head, 16-key tile). A = K rows (16 pos x 4 dims), B cols 0/1 = the 2 GQA query heads.
__global__ void attn_scores_wmma(const float* __restrict__ Kc, const float* __restrict__ qn,
                                 float* __restrict__ scores) {
  int wid  = blockIdx.x * 8 + (threadIdx.x >> 5);
  int lane = threadIdx.x & 31;
  int kh   = wid >> 8;      // 256 tiles per kv head
  int tile = wid & 255;
  int kpos0 = tile * 16;
  int m  = lane & 15;
  int kb = (lane >> 4) * 2;  // K sub-index 0 or 2 (VGPR adds +0/+1)
  const float* Krow = Kc + (((size_t)kh * kCTX) + kpos0 + m) * kDH + kb;
  int j = lane & 15;         // B column = query head within pair (only 0,1 used)
  const float* qrow = qn + (size_t)(kh * 2 + (j < 2 ? j : 0)) * kDH + kb;
  bool jlive = (j < 2);
  v8f c = {};
  for (int d0 = 0; d0 < kDH; d0 += 4) {
    float2 ka = *(const float2*)(Krow + d0);
    v2f a; a.x = ka.x; a.y = ka.y;
    float2 qb = *(const float2*)(qrow + d0);
    v2f b; b.x = jlive ? qb.x : 0.0f; b.y = jlive ? qb.y : 0.0f;
    c = __builtin_amdgcn_wmma_f32_16x16x4_f32(false, a, false, b, (short)0, c, false, false);
  }
  if (jlive) {
    int qh = kh * 2 + j;
    int pbase = kpos0 + (lane >> 4) * 8;  // D rows 0..7 in lanes<16, 8..15 in lanes>=16
    float* srow = scores + (size_t)qh * kCTX + pbase;
#pragma unroll
    for (int r = 0; r < 8; ++r) srow[r] = c[r];
  }
}

// ---------------- softmax(+mask) over CTX per query head ----------------
__global__ void attn_softmax_kernel(float* __restrict__ scores, const float* __restrict__ mask) {
  int h = blockIdx.x;
  float* row = scores + (size_t)h * kCTX;
  float lmax = -3.4e38f;
  for (int i = threadIdx.x; i < kCTX; i += 256) lmax = fmaxf(lmax, row[i] + mask[i]);
  float mx = block_max_256(lmax);
  float lsum = 0.f;
  for (int i = threadIdx.x; i < kCTX; i += 256) {
    float e = expf(row[i] + mask[i] - mx);
    row[i] = e;
    lsum += e;
  }
  float sum = block_sum_256(lsum);
  float inv = 1.0f / sum;
  for (int i = threadIdx.x; i < kCTX; i += 256) row[i] *= inv;
}

// ---------------- attention context via WMMA ----------------
// wave = (kv head, 16-dim chunk, pos-split of 512). A rows 0/1 = softmax weights of the 2 q heads,
// B = V (4 pos x 16 dims). Partials to cpart, reduced deterministically afterwards.
__global__ void attn_ctx_wmma(const float* __restrict__ Vc, const float* __restrict__ probs,
                              float* __restrict__ cpart) {
  int wid  = blockIdx.x * 8 + (threadIdx.x >> 5);
  int lane = threadIdx.x & 31;
  int kh     = wid >> 7;       // 128 waves per kv head
  int rem    = wid & 127;
  int dchunk = rem >> 3;       // 16 chunks of 16 dims
  int split  = rem & 7;        // 8 position splits
  int d0 = dchunk * 16;
  int m  = lane & 15;
  int kb = (lane >> 4) * 2;
  int n  = lane & 15;
  bool mlive = (m < 2);
  const float* prow  = probs + (size_t)(kh * 2 + (mlive ? m : 0)) * kCTX;
  const float* Vbase = Vc + (size_t)kh * kCTX * kDH + d0 + n;
  v8f c = {};
  int p_begin = split * (kCTX / 8);
  for (int p0 = p_begin; p0 < p_begin + kCTX / 8; p0 += 4) {
    float2 pw = *(const float2*)(prow + p0 + kb);
    v2f a; a.x = mlive ? pw.x : 0.0f; a.y = mlive ? pw.y : 0.0f;
    v2f b;
    b.x = Vbase[(size_t)(p0 + kb) * kDH];
    b.y = Vbase[(size_t)(p0 + kb + 1) * kDH];
    c = __builtin_amdgcn_wmma_f32_16x16x4_f32(false, a, false, b, (short)0, c, false, false);
  }
  if (lane < 16) {  // D rows 0,1 live in lanes<16, VGPRs 0,1
    float* base = cpart + (size_t)split * (kH * kDH);
    base[(size_t)(kh * 2 + 0) * kDH + d0 + lane] = c[0];
    base[(size_t)(kh * 2 + 1) * kDH + d0 + lane] = c[1];
  }
}

__global__ void reduce_ctx_kernel(const float* __restrict__ cpart, float* __restrict__ ctx) {
  int i = blockIdx.x * 256 + threadIdx.x;  // H*DH = 4096
  float s = 0.f;
#pragma unroll
  for (int k = 0; k < 8; ++k) s += cpart[(size_t)k * (kH * kDH) + i];
  ctx[i] = s;
}

// ---------------- router: logits, softmax, top-8, scatter ----------------
__global__ void router_kernel(const float* __restrict__ hr, const float* __restrict__ Wr,
                              const float* __restrict__ pes, float* __restrict__ rw) {
  __shared__ float logits[kNE];
  __shared__ float prob[kNE];
  int wv = threadIdx.x >> 5, lane = threadIdx.x & 31;
  for (int e = wv; e < kNE; e += 8) {
    const float4* Wrow = (const float4*)Wr + (size_t)e * (kD / 4);
    const float4* X4 = (const float4*)hr;
    float acc = 0.f;
    for (int i = lane; i < kD / 4; i += 32) {
      float4 w = Wrow[i];
      float4 v = X4[i];
      acc = fmaf(w.x, v.x, fmaf(w.y, v.y, fmaf(w.z, v.z, fmaf(w.w, v.w, acc))));
    }
    acc = wave_sum(acc);
    if (lane == 0) logits[e] = acc;
  }
  __syncthreads();
  if (threadIdx.x == 0) {
    float mx = logits[0];
    for (int e = 1; e < kNE; ++e) mx = fmaxf(mx, logits[e]);
    float sum = 0.f;
    for (int e = 0; e < kNE; ++e) { prob[e] = expf(logits[e] - mx); sum += prob[e]; }
    float inv = 1.0f / sum;
    for (int e = 0; e < kNE; ++e) prob[e] *= inv;
    int   ti[kTOPK];
    float tw[kTOPK];
    bool used[kNE];
    for (int e = 0; e < kNE; ++e) used[e] = false;
    for (int k = 0; k < kTOPK; ++k) {
      int bi = -1; float bv = -1.f;
      for (int e = 0; e < kNE; ++e)
        if (!used[e] && prob[e] > bv) { bv = prob[e]; bi = e; }
      used[bi] = true; ti[k] = bi; tw[k] = bv;
    }
    float s = 0.f;
    for (int k = 0; k < kTOPK; ++k) s += tw[k];
    float is = 1.0f / s;
    for (int e = 0; e < kNE; ++e) rw[e] = 0.0f;
    for (int k = 0; k < kTOPK; ++k) rw[ti[k]] = tw[k] * is * pes[ti[k]];
  }
}

// ---------------- host launcher ----------------
extern "C" void kernel_launch(void* const* d_in, const int* in_sizes, int n_in,
                              void* d_out, int out_size, void* d_ws, size_t ws_size,
                              hipStream_t stream) {
  (void)in_sizes; (void)n_in; (void)out_size; (void)ws_size;
  const float* x            = (const float*)d_in[0];
  const float* cosv         = (const float*)d_in[1];
  const float* sinv         = (const float*)d_in[2];
  const float* k_cache      = (const float*)d_in[3];
  const float* v_cache      = (const float*)d_in[4];
  const float* attn_mask    = (const float*)d_in[5];
  const float* kv_mask      = (const float*)d_in[6];
  const float* input_ln_w   = (const float*)d_in[7];
  const float* q_w          = (const float*)d_in[8];
  const float* k_w          = (const float*)d_in[9];
  const float* v_w          = (const float*)d_in[10];
  const float* o_w          = (const float*)d_in[11];
  const float* q_norm_w     = (const float*)d_in[12];
  const float* k_norm_w     = (const float*)d_in[13];
  const float* post_attn_ln = (const float*)d_in[14];
  const float* pre_ffn_ln   = (const float*)d_in[15];
  const float* dense_gate_w = (const float*)d_in[16];
  const float* dense_up_w   = (const float*)d_in[17];
  const float* dense_down_w = (const float*)d_in[18];
  const float* post_ffn_ln1 = (const float*)d_in[19];
  const float* router_w     = (const float*)d_in[20];
  const float* router_scale = (const float*)d_in[21];
  const float* pexp_scale   = (const float*)d_in[22];
  const float* pre_ffn_ln2  = (const float*)d_in[23];
  const float* pack_gate_w  = (const float*)d_in[24];
  const float* pack_up_w    = (const float*)d_in[25];
  const float* pack_down_w  = (const float*)d_in[26];
  const float* post_ffn_ln2 = (const float*)d_in[27];
  const float* post_ffn_ln  = (const float*)d_in[28];
  const float* layer_scalar = (const float*)d_in[29];

  float* out   = (float*)d_out;
  float* k_new = out + kD;
  float* v_new = k_new + (size_t)kNKV * kCTX * kDH;

  float* ws = (float*)d_ws;
  size_t off = 0;
  auto take = [&](size_t n) -> float* { float* p = ws + off; off += n; return p; };
  float* h      = take(kD);
  float* qraw   = take((size_t)kH * kDH);
  float* kraw   = take((size_t)kNKV * kDH);
  float* vraw   = take((size_t)kNKV * kDH);
  float* qn     = take((size_t)kH * kDH);
  float* kn     = take((size_t)kNKV * kDH);
  float* vn     = take((size_t)kNKV * kDH);
  float* scores = take((size_t)kH * kCTX);
  float* cpart  = take((size_t)8 * kH * kDH);
  float* ctx    = take((size_t)kH * kDH);
  float* atno   = take(kD);
  float* x2     = take(kD);
  float* pre    = take(kD);
  float* tdense = take(kDD);
  float* dense  = take(kD);
  float* h1     = take(kD);
  float* hr     = take(kD);
  float* rw     = take(kNE);
  float* pre2   = take(kD);
  float* hh     = take((size_t)kP * kG * kFF);
  float* moe    = take(kD);
  float* h2     = take(kD);

  // 1) input rmsnorm
  rmsnorm_kernel<<<1, 256, 0, stream>>>(x, input_ln_w, h, kD);
  // 2) QKV projections (streaming GEMV)
  gemv_kernel<<<(kH * kDH) / 8, 256, 0, stream>>>(q_w, h, qraw, kH * kDH, kD / 4);
  gemv_kernel<<<(kNKV * kDH) / 8, 256, 0, stream>>>(k_w, h, kraw, kNKV * kDH, kD / 4);
  gemv_kernel<<<(kNKV * kDH) / 8, 256, 0, stream>>>(v_w, h, vraw, kNKV * kDH, kD / 4);
  // 3) per-head rmsnorm + rope
  qkv_norm_rope_kernel<<<kH + 2 * kNKV, 256, 0, stream>>>(qraw, kraw, vraw, q_norm_w, k_norm_w,
                                                          cosv, sinv, qn, kn, vn);
  // 4) KV cache write-out (also produces k_new / v_new outputs)
  kv_update_kernel<<<(kNKV * kCTX * kDH / 4) / 256, 256, 0, stream>>>(k_cache, kn, kv_mask, k_new);
  kv_update_kernel<<<(kNKV * kCTX * kDH / 4) / 256, 256, 0, stream>>>(v_cache, vn, kv_mask, v_new);
  // 5) attention scores (WMMA f32 16x16x4) + softmax + context (WMMA) + reduce
  attn_scores_wmma<<<(kNKV * (kCTX / 16)) / 8, 256, 0, stream>>>(k_new, qn, scores);
  attn_softmax_kernel<<<kH, 256, 0, stream>>>(scores, attn_mask);
  attn_ctx_wmma<<<(kNKV * 16 * 8) / 8, 256, 0, stream>>>(v_new, scores, cpart);
  reduce_ctx_kernel<<<(kH * kDH) / 256, 256, 0, stream>>>(cpart, ctx);
  // 6) output projection + residual
  gemv_kernel<<<kD / 8, 256, 0, stream>>>(o_w, ctx, atno, kD, (kH * kDH) / 4);
  add_rmsnorm_kernel<<<1, 256, 0, stream>>>(x, atno, post_attn_ln, x2, kD);
  // 7) dense FFN
  rmsnorm_kernel<<<1, 256, 0, stream>>>(x2, pre_ffn_ln, pre, kD);
  gateup_kernel<<<kDD / 8, 256, 0, stream>>>(dense_gate_w, dense_up_w, pre, tdense,
                                             nullptr, 1, kDD, kD / 4);
  gemv_kernel<<<kD / 8, 256, 0, stream>>>(dense_down_w, tdense, dense, kD, kDD / 4);
  rmsnorm_kernel<<<1, 256, 0, stream>>>(dense, post_ffn_ln1, h1, kD);
  // 8) router (must finish before sparse MoE GEMVs)
  router_pre_kernel<<<1, 256, 0, stream>>>(x2, router_scale, hr, kD);
  router_kernel<<<1, 256, 0, stream>>>(hr, router_w, pexp_scale, rw);
  // 9) MoE: only active experts touch weights (rw gating skips 56/64 experts)
  rmsnorm_kernel<<<1, 256, 0, stream>>>(x2, pre_ffn_ln2, pre2, kD);
  gateup_kernel<<<(kP * kG * kFF) / 8, 256, 0, stream>>>(pack_gate_w, pack_up_w, pre2, hh,
                                                         rw, kFF, kP * kG * kFF, kD / 4);
  pack_down_kernel<<<kD / 8, 256, 0, stream>>>(pack_down_w, hh, rw, moe);
  rmsnorm_kernel<<<1, 256, 0, stream>>>(moe, post_ffn_ln2, h2, kD);
  // 10) combine + final residual/scale
  final_kernel<<<1, 256, 0, stream>>>(x2, h1, h2, post_ffn_ln, layer_scalar, out);
}